// GaussianMask_55637006352448
// MI455X (gfx1250) — compile-verified
//
#include <hip/hip_runtime.h>
#include <math.h>

// Problem constants (reference: X = (4, 3, 512, 512) f32, 5x5 window, bw=1.0)
#define HH 512
#define WW 512
#define NPLANES 12          // B*C = 4*3
#define NK 24               // 5*5 - 1 taps
#define TILE 32             // output tile edge per block
#define PADT (TILE + 4)     // padded LDS tile edge (2-halo each side)

#define GLOBAL_AS __attribute__((address_space(1)))
#define LDS_AS    __attribute__((address_space(3)))

typedef __attribute__((ext_vector_type(4))) float v4f;

// exp(-d^2/2) = exp2(-d^2 * 0.5*log2(e))
#define NEG_HALF_LOG2E (-0.72134752044448170368f)

__global__ __launch_bounds__(256)
void gauss_mask_kernel(const float* __restrict__ X, float* __restrict__ out) {
    __shared__ float tile[PADT][PADT];   // 36*36*4 = 5184 B

    const int tw    = blockIdx.x * TILE;     // tile origin (w)
    const int th    = blockIdx.y * TILE;     // tile origin (h)
    const int plane = blockIdx.z;            // 0..11  (b*C + c)
    const float* __restrict__ Xp = X + (size_t)plane * (HH * WW);

    const int tid = threadIdx.x;             // 0..255

    // ---- Stage padded 36x36 tile into LDS ----------------------------------
    // In-bounds elements: async global->LDS DMA (ASYNCcnt path, no VGPR staging).
    // Out-of-bounds (zero padding): direct DS store of 0.
    for (int e = tid; e < PADT * PADT; e += 256) {
        const int py = e / PADT;
        const int px = e - py * PADT;
        const int gh = th + py - 2;
        const int gw = tw + px - 2;
        if (gh >= 0 && gh < HH && gw >= 0 && gw < WW) {
#if __has_builtin(__builtin_amdgcn_global_load_async_to_lds_b32)
            __builtin_amdgcn_global_load_async_to_lds_b32(
                (GLOBAL_AS int*)(Xp + (size_t)gh * WW + gw),
                (LDS_AS int*)&tile[py][px],
                /*offset=*/0, /*cpol=*/0);
#else
            tile[py][px] = Xp[(size_t)gh * WW + gw];
#endif
        } else {
            tile[py][px] = 0.0f;
        }
    }
#if __has_builtin(__builtin_amdgcn_global_load_async_to_lds_b32)
    asm volatile("s_wait_asynccnt 0x0" ::: "memory");
#endif
    __syncthreads();

    // ---- Compute: each thread owns 4 contiguous pixels of one row ----------
    const int r  = tid >> 3;           // 0..31 row within tile
    const int c0 = (tid & 7) * 4;      // 0,4,...,28 col group within tile

    const float xi0 = tile[r + 2][c0 + 2 + 0];
    const float xi1 = tile[r + 2][c0 + 2 + 1];
    const float xi2 = tile[r + 2][c0 + 2 + 2];
    const float xi3 = tile[r + 2][c0 + 2 + 3];

    // out[((plane)*24 + k)*H*W + h*W + w]
    const size_t planeSz = (size_t)HH * WW;
    const size_t obase   = (size_t)plane * NK * planeSz
                         + (size_t)(th + r) * WW + (tw + c0);
    float* __restrict__ obp = out + obase;

#pragma unroll
    for (int k = 0; k < NK; ++k) {
        const int idx = (k < NK / 2) ? k : k + 1;   // skip center tap
        const int dr  = idx / 5;
        const int dc  = idx % 5;

        const float d0 = tile[r + dr][c0 + dc + 0] - xi0;
        const float d1 = tile[r + dr][c0 + dc + 1] - xi1;
        const float d2 = tile[r + dr][c0 + dc + 2] - xi2;
        const float d3 = tile[r + dr][c0 + dc + 3] - xi3;

        v4f v;
        v.x = __builtin_amdgcn_exp2f(d0 * d0 * NEG_HALF_LOG2E);
        v.y = __builtin_amdgcn_exp2f(d1 * d1 * NEG_HALF_LOG2E);
        v.z = __builtin_amdgcn_exp2f(d2 * d2 * NEG_HALF_LOG2E);
        v.w = __builtin_amdgcn_exp2f(d3 * d3 * NEG_HALF_LOG2E);

        // Streaming 302 MB output > 192 MB L2: non-temporal 128-bit store.
        __builtin_nontemporal_store(v, (v4f*)(obp + (size_t)k * planeSz));
    }
}

extern "C" void kernel_launch(void* const* d_in, const int* in_sizes, int n_in,
                              void* d_out, int out_size, void* d_ws, size_t ws_size,
                              hipStream_t stream) {
    const float* X = (const float*)d_in[0];
    float* out     = (float*)d_out;

    dim3 grid(WW / TILE, HH / TILE, NPLANES);   // 16 x 16 x 12
    dim3 block(256);
    gauss_mask_kernel<<<grid, block, 0, stream>>>(X, out);
}